// ESPRIT_39805756899660
// MI455X (gfx1250) — compile-verified
//
#include <hip/hip_runtime.h>
#include <math.h>

typedef __attribute__((ext_vector_type(2))) float v2f;
typedef __attribute__((ext_vector_type(8))) float v8f;

#if defined(__has_builtin)
#if __has_builtin(__builtin_amdgcn_global_load_async_to_lds_b128) && \
    __has_builtin(__builtin_amdgcn_s_wait_asynccnt)
#define HAVE_ASYNC_LDS 1
#endif
#endif

namespace {
constexpr int   kNfft   = 512;
constexpr int   kHop    = 256;
constexpr int   kMic    = 4;
constexpr int   kFrames = 4;                      // EST_NUM snapshots
constexpr int   kBin0   = 13;                     // first rfft bin with f >= 400
constexpr int   kNbin   = 19;                     // bins 13..31 (400 <= f < 1000)
constexpr int   kSamp0  = 120 * kHop - kNfft / 2; // 30464 = first raw sample needed
constexpr int   kRows   = 16;                     // (mic,frame) pairs
constexpr int   kCols   = 48;                     // 19 bins * {re,im} padded to 3 WMMA tiles
constexpr int   kXN     = (3 * kHop + kNfft) * kMic; // 5120 staged floats (20 KB)
constexpr float kPi     = 3.14159265358979323846f;
}

struct C2 { float x, y; };
__device__ inline C2 cadd (C2 a, C2 b){ return {a.x+b.x, a.y+b.y}; }
__device__ inline C2 csub (C2 a, C2 b){ return {a.x-b.x, a.y-b.y}; }
__device__ inline C2 cmul (C2 a, C2 b){ return {a.x*b.x-a.y*b.y, a.x*b.y+a.y*b.x}; }
__device__ inline C2 cjmul(C2 a, C2 b){ return {a.x*b.x+a.y*b.y, a.x*b.y-a.y*b.x}; } // conj(a)*b
__device__ inline C2 cscale(C2 a, float s){ return {a.x*s, a.y*s}; }

__global__ __launch_bounds__(32)
void esprit_fused_kernel(const float* __restrict__ x, float* __restrict__ out) {
  __shared__ float xs[kXN];          // staged raw samples x[30464..31743][0..3]
  __shared__ float ctab[kNfft];      // cos(2*pi*i/512)
  __shared__ float wtab[kNfft];      // sqrt(hann)[k] = sin(pi*k/512)
  __shared__ float D[kRows * kCols]; // 16 x 48 DFT result

  const int lane  = threadIdx.x;     // single wave32
  const int half  = lane >> 4;
  const int l16   = lane & 15;
  const int kbase = 2 * half;        // K placement per ISA 7.12.2 (f32 16x4 layout)

  // ---- Phase 0a: kick off the 20 KB x staging (async LDS DMA when available) ----
  const float* xg = x + (long)kSamp0 * kMic;
#ifdef HAVE_ASYNC_LDS
  {
    typedef int v4i_ __attribute__((vector_size(4 * sizeof(int))));
    typedef __attribute__((address_space(1))) v4i_* as1_v4i;
    typedef __attribute__((address_space(3))) v4i_* as3_v4i;
    for (int i = lane; i < kXN / 4; i += 32) {
      __builtin_amdgcn_global_load_async_to_lds_b128(
          (as1_v4i)(xg + 4 * i), (as3_v4i)(xs + 4 * i), 0, 0);
    }
  }
#else
  __builtin_prefetch(xg, 0, 1);
#endif

  // ---- Phase 0b: twiddle + window tables (exact reduced args -> fast sincos path)
  for (int i = lane; i < kNfft; i += 32) {
    float s, c;
    sincosf((2.0f * kPi / (float)kNfft) * (float)i, &s, &c);
    ctab[i] = c;
    wtab[i] = sinf((kPi / (float)kNfft) * (float)i);
  }

#ifdef HAVE_ASYNC_LDS
  __builtin_amdgcn_s_wait_asynccnt(0);
#else
  {
    const float4* g4 = (const float4*)xg;
    float4*       l4 = (float4*)xs;
    for (int i = lane; i < kXN / 4; i += 32) l4[i] = g4[i];
  }
#endif
  __syncthreads();

  // ---- Phase 1: (16 x 512) x (512 x 48) DFT GEMM on V_WMMA_F32_16X16X4_F32 ----
  // A row M = l16 -> (mic m, frame t): A[M][k] = xs[(t*HOP + k)*4 + m] * wtab[k]
  const int m = l16 >> 2;
  const int t = l16 & 3;
  const int abase = t * kHop * kMic + m;

  // This lane's output column per tile: twiddle = ctab[(bin*k + off) & 511],
  // off = 0 (real part, cos) or 128 (imag part, -sin = cos(theta + pi/2)).
  // Padded columns 38..47 compute garbage that is never read.
  int bo[3], ofs[3];
  #pragma unroll
  for (int T = 0; T < 3; ++T) {
    const int c = T * 16 + l16;
    bo[T]  = kBin0 + (c >> 1);
    ofs[T] = (c & 1) ? (kNfft / 4) : 0;   // +128 index = +pi/2 phase
  }

  v8f acc0 = {}, acc1 = {}, acc2 = {};
  for (int s = 0; s < kNfft / 4; ++s) {
    const int k0 = 4 * s + kbase, k1 = k0 + 1;
    v2f a, b0, b1, b2;
    a.x  = xs[abase + kMic * k0] * wtab[k0];
    a.y  = xs[abase + kMic * k1] * wtab[k1];
    b0.x = ctab[(bo[0] * k0 + ofs[0]) & (kNfft - 1)];
    b0.y = ctab[(bo[0] * k1 + ofs[0]) & (kNfft - 1)];
    b1.x = ctab[(bo[1] * k0 + ofs[1]) & (kNfft - 1)];
    b1.y = ctab[(bo[1] * k1 + ofs[1]) & (kNfft - 1)];
    b2.x = ctab[(bo[2] * k0 + ofs[2]) & (kNfft - 1)];
    b2.y = ctab[(bo[2] * k1 + ofs[2]) & (kNfft - 1)];
    acc0 = __builtin_amdgcn_wmma_f32_16x16x4_f32(false, a, false, b0, (short)0, acc0, false, false);
    acc1 = __builtin_amdgcn_wmma_f32_16x16x4_f32(false, a, false, b1, (short)0, acc1, false, false);
    acc2 = __builtin_amdgcn_wmma_f32_16x16x4_f32(false, a, false, b2, (short)0, acc2, false, false);
  }

  // C/D layout: VGPR v holds (M = v + 8*half, N = l16)
  #pragma unroll
  for (int v = 0; v < 8; ++v) {
    const int row = v + 8 * half;
    D[row * kCols +  0 + l16] = acc0[v];
    D[row * kCols + 16 + l16] = acc1[v];
    D[row * kCols + 32 + l16] = acc2[v];
  }
  __syncthreads();

  // ---- Phase 2: per-bin ESPRIT, one lane per frequency bin ----
  const bool active = lane < kNbin;
  const int  j      = active ? lane : 0;

  C2 X[kMic][kFrames];
  #pragma unroll
  for (int mm = 0; mm < kMic; ++mm)
    #pragma unroll
    for (int tt = 0; tt < kFrames; ++tt) {
      const int r = mm * kFrames + tt;
      X[mm][tt] = { D[r * kCols + 2 * j], D[r * kCols + 2 * j + 1] };
    }

  // R = Xs Xs^H / 4 (4x4 Hermitian)
  C2 R[kMic][kMic];
  for (int a_ = 0; a_ < 4; ++a_)
    for (int b_ = 0; b_ < 4; ++b_) {
      C2 s = {0.f, 0.f};
      for (int tt = 0; tt < 4; ++tt) s = cadd(s, cjmul(X[b_][tt], X[a_][tt]));
      R[a_][b_] = cscale(s, 0.25f);
    }

  // Orthogonal iteration -> basis of the top-2 eigenspace (signal subspace).
  // ESPRIT eigenvalues are invariant under ANY invertible basis change of it.
  C2 V[4][2];
  for (int i = 0; i < 4; ++i) {
    V[i][0] = { (i == 0) ? 1.f : 0.f, 0.f };
    V[i][1] = { (i == 1) ? 1.f : 0.f, 0.f };
  }
  for (int it = 0; it < 10; ++it) {
    C2 W[4][2];
    for (int i = 0; i < 4; ++i)
      for (int c = 0; c < 2; ++c) {
        C2 s = {0.f, 0.f};
        for (int n = 0; n < 4; ++n) s = cadd(s, cmul(R[i][n], V[n][c]));
        W[i][c] = s;
      }
    float n0 = 1e-30f;
    for (int i = 0; i < 4; ++i) n0 += W[i][0].x * W[i][0].x + W[i][0].y * W[i][0].y;
    const float i0 = rsqrtf(n0);
    for (int i = 0; i < 4; ++i) W[i][0] = cscale(W[i][0], i0);
    C2 d = {0.f, 0.f};
    for (int i = 0; i < 4; ++i) d = cadd(d, cjmul(W[i][0], W[i][1]));
    for (int i = 0; i < 4; ++i) W[i][1] = csub(W[i][1], cmul(d, W[i][0]));
    float n1 = 1e-30f;
    for (int i = 0; i < 4; ++i) n1 += W[i][1].x * W[i][1].x + W[i][1].y * W[i][1].y;
    const float i1 = rsqrtf(n1);
    for (int i = 0; i < 4; ++i) { V[i][0] = W[i][0]; V[i][1] = cscale(W[i][1], i1); }
  }

  // s0 = rows 0..2, s1 = rows 1..3; A = s0^H s0, B = s0^H s1 (2x2)
  C2 A00={0,0},A01={0,0},A10={0,0},A11={0,0};
  C2 B00={0,0},B01={0,0},B10={0,0},B11={0,0};
  for (int i = 0; i < 3; ++i) {
    A00 = cadd(A00, cjmul(V[i][0], V[i][0]));
    A01 = cadd(A01, cjmul(V[i][0], V[i][1]));
    A10 = cadd(A10, cjmul(V[i][1], V[i][0]));
    A11 = cadd(A11, cjmul(V[i][1], V[i][1]));
    B00 = cadd(B00, cjmul(V[i][0], V[i+1][0]));
    B01 = cadd(B01, cjmul(V[i][0], V[i+1][1]));
    B10 = cadd(B10, cjmul(V[i][1], V[i+1][0]));
    B11 = cadd(B11, cjmul(V[i][1], V[i+1][1]));
  }
  // phi = A^{-1} B
  C2 detA = csub(cmul(A00, A11), cmul(A01, A10));
  const float dmag = detA.x * detA.x + detA.y * detA.y + 1e-38f;
  C2 idet = { detA.x / dmag, -detA.y / dmag };
  C2 p00 = cmul(idet, csub(cmul(A11, B00), cmul(A01, B10)));
  C2 p01 = cmul(idet, csub(cmul(A11, B01), cmul(A01, B11)));
  C2 p10 = cmul(idet, csub(cmul(A00, B10), cmul(A10, B00)));
  C2 p11 = cmul(idet, csub(cmul(A00, B11), cmul(A10, B01)));

  // Eigenvalues of phi via quadratic formula with principal complex sqrt
  C2 tr = cadd(p00, p11);
  C2 dt = csub(cmul(p00, p11), cmul(p01, p10));
  C2 z  = csub(cmul(tr, tr), cscale(dt, 4.0f));
  const float rm = sqrtf(z.x * z.x + z.y * z.y);
  C2 disc;
  disc.x = sqrtf(fmaxf(0.5f * (rm + z.x), 0.0f));
  disc.y = copysignf(sqrtf(fmaxf(0.5f * (rm - z.x), 0.0f)), z.y);
  C2 l1 = cscale(cadd(tr, disc), 0.5f);
  C2 l2 = cscale(csub(tr, disc), 0.5f);

  const float freq = 31.25f * (float)(kBin0 + j);          // FS/NFFT * bin
  const float sc   = 343.0f / (2.0f * kPi * freq * 0.08f); // c / (2 pi f d)
  const float r2d  = 180.0f / kPi;
  float angsum = r2d * (asinf(atan2f(l1.y, l1.x) * sc) +
                        asinf(atan2f(l2.y, l2.x) * sc));
  float local = active ? angsum : 0.0f;

  #pragma unroll
  for (int off = 16; off > 0; off >>= 1)
    local += __shfl_down(local, off, 32);
  if (lane == 0) out[0] = local / (float)(kNbin * 2);
}

extern "C" void kernel_launch(void* const* d_in, const int* in_sizes, int n_in,
                              void* d_out, int out_size, void* d_ws, size_t ws_size,
                              hipStream_t stream) {
  (void)in_sizes; (void)n_in; (void)out_size; (void)d_ws; (void)ws_size;
  const float* x = (const float*)d_in[0];   // (1920000, 4) float32 row-major
  float* out = (float*)d_out;               // scalar float32
  esprit_fused_kernel<<<1, 32, 0, stream>>>(x, out);
}